// Interpolate_37744172597322
// MI455X (gfx1250) — compile-verified
//
#include <hip/hip_runtime.h>

typedef __attribute__((ext_vector_type(2))) float v2f;
typedef __attribute__((ext_vector_type(8))) float v8f;

#define BATCH 8
#define NQ    8192
#define MP    2048
#define CH    256
#define TILES (MP / 16)       // 128 WMMA column tiles
#define ROWS_PER_WAVE 16
#define WAVES_PER_BLOCK 8
#define ROWS_PER_BLOCK (ROWS_PER_WAVE * WAVES_PER_BLOCK)  // 128
#define EPSF 1e-10f

#define KSCALE 131072.0f          // 2^17, folded into A operand
#define KBIAS  (8.0f * KSCALE)    // dot in [-6,3] -> dot+8 >= 2 > 0, folded into C
#define IDXMASK 0x7FFu            // 11 bits of column index in key LSBs

// median of three u32 -- min/max tree pattern-matched to v_med3_u32
__device__ __forceinline__ unsigned umed3(unsigned a, unsigned b, unsigned c) {
    unsigned mn = a < b ? a : b;
    unsigned mx = a < b ? b : a;
    unsigned t  = mx < c ? mx : c;
    return mn > t ? mn : t;
}

// insert v into ascending sorted triple of packed keys: 1 min + 2 med3
__device__ __forceinline__ void insert3u(unsigned& s0, unsigned& s1, unsigned& s2,
                                         unsigned v) {
    unsigned n0 = v < s0 ? v : s0;
    unsigned n1 = umed3(v, s0, s1);
    unsigned n2 = umed3(v, s1, s2);
    s0 = n0; s1 = n1; s2 = n2;
}

__global__ __launch_bounds__(256) void three_interp_wmma_kernel(
    const float* __restrict__ features,   // [B, M, C]
    const float* __restrict__ xyz1,       // [B, N, 3]
    const float* __restrict__ xyz2,       // [B, M, 3]
    float* __restrict__ out)              // [B, N, C]
{
    // (x2, y2, z2, |p2|^2) per reference point. Doubles as the WMMA B operand
    // source: lane<16 reads .xy (K0,K1), lane>=16 reads .zw (K2,K3).
    __shared__ float4 sP[MP];             // 32 KB

    const int tid  = threadIdx.x;
    const int lane = tid & 31;
    const int wave = tid >> 5;
    const int b    = blockIdx.y;
    const int n0   = blockIdx.x * ROWS_PER_BLOCK + wave * ROWS_PER_WAVE;

    // ---- cooperative LDS fill for this batch ----
    const float* x2base = xyz2 + (size_t)b * MP * 3;
    for (int col = tid; col < MP; col += 256) {
        const float* p = x2base + (size_t)col * 3;
        float x = p[0], y = p[1], z = p[2];
        sP[col] = make_float4(x, y, z, x * x + y * y + z * z);
    }
    __syncthreads();

    // ---- A operand: rows n0..n0+15, K = 2^17 * [-2x, -2y, -2z, 1] ----
    const int row16 = lane & 15;
    const int khalf = lane >> 4;
    const float* pa = xyz1 + ((size_t)b * NQ + n0 + row16) * 3;
    const float ax = pa[0], ay = pa[1], az = pa[2];
    const float sq1 = ax * ax + ay * ay + az * az;
    v2f aop;
    aop.x = khalf ? (-2.0f * az * KSCALE) : (-2.0f * ax * KSCALE);
    aop.y = khalf ? KSCALE                : (-2.0f * ay * KSCALE);

    // C = bias so D = 2^17*(dot+8) is positive and fits 21 bits
    const v8f cbias = {KBIAS, KBIAS, KBIAS, KBIAS, KBIAS, KBIAS, KBIAS, KBIAS};

    // ---- packed-key top-3 state per D-matrix slot ----
    unsigned s0[8], s1[8], s2[8];
#pragma unroll
    for (int r = 0; r < 8; ++r) { s0[r] = s1[r] = s2[r] = 0xFFFFFFFFu; }

    // per-lane B fetch position within a tile (float2 units)
    const float2* sp2 = (const float2*)sP;
    const int bidx = row16 * 2 + khalf;

    // ---- distance GEMM: 128 x v_wmma_f32_16x16x4_f32, LDS load pipelined ----
    float2 bv = sp2[bidx];                // tile 0 operand
#pragma unroll 2
    for (int t = 0; t < TILES; ++t) {
        v2f bop;
        bop.x = bv.x;
        bop.y = bv.y;
        float2 nxt = sp2[(((t + 1) & (TILES - 1)) << 5) + bidx];
        v8f c = __builtin_amdgcn_wmma_f32_16x16x4_f32(
            false, aop, false, bop, (short)0, cbias, false, false);
        const unsigned m = (unsigned)(t * 16 + row16);
#pragma unroll
        for (int r = 0; r < 8; ++r) {
            // v_cvt_u32_f32 + v_lshl_or_b32
            unsigned key = (((unsigned)c[r]) << 11) | m;
            insert3u(s0[r], s1[r], s2[r], key);
        }
        bv = nxt;
    }

    // ---- merge sorted key-triples across the 16 lanes of each half ----
#pragma unroll
    for (int s = 0; s < 4; ++s) {
        const int mask = 1 << s;
#pragma unroll
        for (int r = 0; r < 8; ++r) {
            unsigned e0 = (unsigned)__shfl_xor((int)s0[r], mask, 32);
            unsigned e1 = (unsigned)__shfl_xor((int)s1[r], mask, 32);
            unsigned e2 = (unsigned)__shfl_xor((int)s2[r], mask, 32);
            insert3u(s0[r], s1[r], s2[r], e0);
            insert3u(s0[r], s1[r], s2[r], e1);
            insert3u(s0[r], s1[r], s2[r], e2);
        }
    }

    // ---- weighted gather: row by row, lane covers 8 channels ----
    const float* featB = features + (size_t)b * MP * CH;
    float* outB = out + ((size_t)b * NQ + n0) * CH;
    const int cbase = lane * 8;

#pragma unroll
    for (int row = 0; row < ROWS_PER_WAVE; ++row) {
        const int r   = row & 7;
        const int src = (row < 8) ? 0 : 16;   // rows 0-7 in lanes<16, 8-15 in lanes>=16
        int j0 = __shfl((int)s0[r], src, 32) & IDXMASK;
        int j1 = __shfl((int)s1[r], src, 32) & IDXMASK;
        int j2 = __shfl((int)s2[r], src, 32) & IDXMASK;

        // broadcast this query row's coordinates (lane `row` loaded them)
        float rx  = __shfl(ax,  row, 32);
        float ry  = __shfl(ay,  row, 32);
        float rz  = __shfl(az,  row, 32);
        float s1v = __shfl(sq1, row, 32);

        // exact f32 distances, same formula as reference: sq1 + sq2 - 2*cross
        float4 p0 = sP[j0], p1 = sP[j1], p2 = sP[j2];
        float q0 = fmaxf(s1v + p0.w - 2.0f * (rx * p0.x + ry * p0.y + rz * p0.z), EPSF);
        float q1 = fmaxf(s1v + p1.w - 2.0f * (rx * p1.x + ry * p1.y + rz * p1.z), EPSF);
        float q2 = fmaxf(s1v + p2.w - 2.0f * (rx * p2.x + ry * p2.y + rz * p2.z), EPSF);
        float r0 = 1.0f / q0, r1 = 1.0f / q1, r2 = 1.0f / q2;
        float rs = 1.0f / (r0 + r1 + r2);
        float w0 = r0 * rs, w1 = r1 * rs, w2 = r2 * rs;

        const float4* f0 = (const float4*)(featB + (size_t)j0 * CH + cbase);
        const float4* f1 = (const float4*)(featB + (size_t)j1 * CH + cbase);
        const float4* f2 = (const float4*)(featB + (size_t)j2 * CH + cbase);
        float4 a0 = f0[0], a1 = f0[1];
        float4 b0 = f1[0], b1 = f1[1];
        float4 c0 = f2[0], c1 = f2[1];

        float4 o0, o1;
        o0.x = w0 * a0.x + w1 * b0.x + w2 * c0.x;
        o0.y = w0 * a0.y + w1 * b0.y + w2 * c0.y;
        o0.z = w0 * a0.z + w1 * b0.z + w2 * c0.z;
        o0.w = w0 * a0.w + w1 * b0.w + w2 * c0.w;
        o1.x = w0 * a1.x + w1 * b1.x + w2 * c1.x;
        o1.y = w0 * a1.y + w1 * b1.y + w2 * c1.y;
        o1.z = w0 * a1.z + w1 * b1.z + w2 * c1.z;
        o1.w = w0 * a1.w + w1 * b1.w + w2 * c1.w;

        float* op = outB + (size_t)row * CH + cbase;
        *(float4*)(op)     = o0;
        *(float4*)(op + 4) = o1;
    }
}

extern "C" void kernel_launch(void* const* d_in, const int* in_sizes, int n_in,
                              void* d_out, int out_size, void* d_ws, size_t ws_size,
                              hipStream_t stream) {
    const float* features = (const float*)d_in[0];   // [8, 2048, 256]
    const float* xyz1     = (const float*)d_in[1];   // [8, 8192, 3]
    const float* xyz2     = (const float*)d_in[2];   // [8, 2048, 3]
    float* out            = (float*)d_out;           // [8, 8192, 256]

    dim3 grid(NQ / ROWS_PER_BLOCK, BATCH);           // (64, 8)
    dim3 block(256);
    three_interp_wmma_kernel<<<grid, block, 0, stream>>>(features, xyz1, xyz2, out);
}